// RegionLoss_2817498546383
// MI455X (gfx1250) — compile-verified
//
#include <hip/hip_runtime.h>
#include <math.h>

// ---------------- problem constants (fixed by the reference) ----------------
#define NB 16
#define NA 5
#define NC 20
#define NH 128
#define NW 128
#define MAXB 50
#define HW (NH * NW)              // 16384
#define NLOC (NB * NA * HW)       // 1,310,720 (rows of the cls matrix too)
#define NV4 (NLOC / 4)            // 327,680 float4 groups (dense conf pass)
#define SLAB4 (HW / 4)            // 4096 float4 per (b,a) conf slab
#define LOG20 2.9957323f

// ---------------- workspace layout (floats) ----------------
// records: [live, a, j, i, tx, ty, tw, th, tconf, tcls, pad, pad] x 800
#define REC_STRIDE 12
#define NREC (NB * MAXB)                    // 800
#define ACC_BASE (NREC * REC_STRIDE)        // 9600: [0]=nGT, [1]=nMask
#define PART_BASE (ACC_BASE + 16)           // 9616: dense per-block partials
#define PART_MAX 1024
#define CORR_BASE (PART_BASE + PART_MAX)    // 10640: [lx,ly,lw,lh,confc,clsc]
// total ~42.6 KB of d_ws needed.

__device__ __constant__ float c_aw[NA] = {1.3221f, 3.19275f, 5.05587f, 9.47112f, 11.2364f};
__device__ __constant__ float c_ah[NA] = {1.73145f, 4.00944f, 8.09892f, 4.84053f, 10.0071f};

__device__ __forceinline__ float sigmoidf_(float v) { return 1.0f / (1.0f + __expf(-v)); }

// ---------------- CDNA5 WMMA wave reduction (exact fp32 ones-matmul) -------
// D = A x B, A(16x4) holds (s_lane, 0) per lane, B(4x16) = all ones
// => D[m,n] = s_m + s_{m+16}; each lane sums its 8 D regs, two shfls finish.
typedef float v2f_t __attribute__((ext_vector_type(2)));
typedef float v8f_t __attribute__((ext_vector_type(8)));

__device__ __forceinline__ float wave_sum_wmma(float s) {
  v2f_t a; a[0] = s;    a[1] = 0.0f;
  v2f_t b; b[0] = 1.0f; b[1] = 1.0f;   // all-ones B: layout-independent
  v8f_t c = {};
  v8f_t d = __builtin_amdgcn_wmma_f32_16x16x4_f32(
      /*neg_a=*/false, a, /*neg_b=*/false, b,
      /*c_mod=*/(short)0, c, /*reuse_a=*/false, /*reuse_b=*/false);
  float t = d[0] + d[1] + d[2] + d[3] + d[4] + d[5] + d[6] + d[7];
  // lanes 0-15 hold sum(s_0..7 + s_16..23), lanes 16-31 hold the rest
  return __shfl(t, 0, 32) + __shfl(t, 16, 32);
}

// ---------------- kernel 1: build_targets (anchor match + scatter dedup) ----
__global__ void bt_kernel(const float* __restrict__ tgt, float* __restrict__ ws) {
  if (threadIdx.x == 0)
    for (int k = 0; k < 16; ++k) ws[ACC_BASE + k] = 0.0f;
  __syncthreads();
  int b = threadIdx.x;
  if (b >= NB) return;

  const float* t = tgt + b * MAXB * 5;
  int bn[MAXB], jj[MAXB], ii[MAXB];
  int nvalid = 0;
  for (int k = 0; k < MAXB; ++k) {
    float cls = t[5 * k + 0], gx = t[5 * k + 1], gy = t[5 * k + 2];
    float gw  = t[5 * k + 3], gh = t[5 * k + 4];
    if (gx == 0.0f) break;                       // cumprod(valid) semantics
    float w = gw * NW, h = gh * NH;
    int i = (int)(gx * NW); i = min(max(i, 0), NW - 1);
    int j = (int)(gy * NH); j = min(max(j, 0), NH - 1);
    int best = 0; float bestiou = -1.0f;
    for (int a = 0; a < NA; ++a) {
      float inter = fminf(w, c_aw[a]) * fminf(h, c_ah[a]);
      float uni   = w * h + c_aw[a] * c_ah[a] - inter;
      float iou   = inter / uni;
      if (iou > bestiou) { bestiou = iou; best = a; }   // argmax = first max
    }
    bn[nvalid] = best; jj[nvalid] = j; ii[nvalid] = i;
    float* r = ws + (size_t)(b * MAXB + nvalid) * REC_STRIDE;
    r[1] = (float)best; r[2] = (float)j; r[3] = (float)i;
    r[4] = gx * NW - (float)i;
    r[5] = gy * NH - (float)j;
    r[6] = logf(fmaxf(w, 1e-12f) / c_aw[best]);
    r[7] = logf(fmaxf(h, 1e-12f) / c_ah[best]);
    r[8] = bestiou;
    r[9] = cls;
    ++nvalid;
  }
  // last-write-wins dedup of the scatter: record k dies if a later valid box
  // hits the same (anchor, j, i) cell.
  int nmask = 0;
  for (int k = 0; k < MAXB; ++k) {
    float* r = ws + (size_t)(b * MAXB + k) * REC_STRIDE;
    int live = 0;
    if (k < nvalid) {
      live = 1;
      for (int k2 = k + 1; k2 < nvalid; ++k2)
        if (bn[k2] == bn[k] && jj[k2] == jj[k] && ii[k2] == ii[k]) { live = 0; break; }
    } else {
      r[1] = r[2] = r[3] = r[4] = r[5] = r[6] = r[7] = r[8] = r[9] = 0.0f;
    }
    r[0] = (float)live;
    nmask += live;
  }
  atomicAdd(ws + ACC_BASE + 0, (float)nvalid);   // exact small-int adds
  atomicAdd(ws + ACC_BASE + 1, (float)nmask);
}

// ---------------- kernel 2: dense Sum(sigmoid(conf)^2), 5.24 MB traffic -----
__global__ void conf_dense_kernel(const float* __restrict__ out, float* __restrict__ ws) {
  __shared__ float wsum[8];
  float s = 0.0f;
  for (int idx = blockIdx.x * blockDim.x + threadIdx.x; idx < NV4;
       idx += gridDim.x * blockDim.x) {
    int b   = idx / (NA * SLAB4);
    int rem = idx - b * (NA * SLAB4);
    int a   = rem / SLAB4;
    int hw4 = rem - a * SLAB4;
    const float4* p = (const float4*)(out + (size_t)(b * 125 + a * 25 + 4) * HW) + hw4;
    float4 v = *p;
    float c0 = sigmoidf_(v.x), c1 = sigmoidf_(v.y);
    float c2 = sigmoidf_(v.z), c3 = sigmoidf_(v.w);
    s += c0 * c0 + c1 * c1 + c2 * c2 + c3 * c3;
  }
  // all lanes reconverged here -> EXEC all ones for WMMA
  float w = wave_sum_wmma(s);
  int wave = threadIdx.x >> 5, lane = threadIdx.x & 31;
  if (lane == 0) wsum[wave] = w;
  __syncthreads();
  if (threadIdx.x == 0) {
    float tot = 0.0f;
    for (int k = 0; k < (int)(blockDim.x >> 5); ++k) tot += wsum[k];
    ws[PART_BASE + blockIdx.x] = tot;   // unconditional write; no init needed
  }
}

// ---------------- kernel 3: sparse corrections at <=800 masked cells --------
__global__ void corr_kernel(const float* __restrict__ out, float* __restrict__ ws) {
  __shared__ float sdata[256];
  float acc[6] = {0, 0, 0, 0, 0, 0};
  for (int rix = threadIdx.x; rix < NREC; rix += blockDim.x) {
    const float* r = ws + (size_t)rix * REC_STRIDE;
    if (r[0] == 0.0f) continue;
    int b = rix / MAXB;
    int a = (int)r[1], j = (int)r[2], i = (int)r[3];
    float tx = r[4], ty = r[5], tw = r[6], th = r[7], tconf = r[8];
    int tc = (int)r[9]; tc = min(max(tc, 0), NC - 1);
    size_t base = ((size_t)b * 125 + (size_t)a * 25) * HW + (size_t)j * NW + i;
    float o0 = out[base + 0 * HW], o1 = out[base + 1 * HW];
    float o2 = out[base + 2 * HW], o3 = out[base + 3 * HW];
    float conf = sigmoidf_(out[base + 4 * HW]);
    float dx = sigmoidf_(o0) - tx, dy = sigmoidf_(o1) - ty;
    float dw = o2 - tw,            dh = o3 - th;
    acc[0] += dx * dx; acc[1] += dy * dy; acc[2] += dw * dw; acc[3] += dh * dh;
    float dc = 5.0f * (conf - tconf);
    acc[4] += dc * dc - conf * conf;           // replace weight-1 base term
    // cls row = channels 4..23 (reference's 4:4+nC slice), log-softmax NLL
    float vals[NC], m = -3.4e38f;
    for (int c = 0; c < NC; ++c) { vals[c] = out[base + (size_t)(4 + c) * HW]; m = fmaxf(m, vals[c]); }
    float sum = 0.0f;
    for (int c = 0; c < NC; ++c) sum += __expf(vals[c] - m);
    float lse = m + logf(sum);
    acc[5] += (lse - vals[tc]) - LOG20;        // replace base log(20) term
  }
  for (int comp = 0; comp < 6; ++comp) {       // deterministic tree reduce
    sdata[threadIdx.x] = acc[comp];
    __syncthreads();
    for (int off = blockDim.x >> 1; off > 0; off >>= 1) {
      if (threadIdx.x < off) sdata[threadIdx.x] += sdata[threadIdx.x + off];
      __syncthreads();
    }
    if (threadIdx.x == 0) ws[CORR_BASE + comp] = sdata[0];
    __syncthreads();
  }
}

// ---------------- kernel 4: finalize scalar ---------------------------------
__global__ void finalize_kernel(const float* __restrict__ ws, float* __restrict__ o, int nblk) {
  if (threadIdx.x == 0 && blockIdx.x == 0) {
    float dense = 0.0f;
    for (int k = 0; k < nblk; ++k) dense += ws[PART_BASE + k];
    float nGT_f = fmaxf(ws[ACC_BASE + 0], 1.0f);
    float lx = ws[CORR_BASE + 0], ly = ws[CORR_BASE + 1];
    float lw = ws[CORR_BASE + 2], lh = ws[CORR_BASE + 3];
    float loss_conf = dense + ws[CORR_BASE + 4];
    float loss_cls  = (float)NLOC * LOG20 + ws[CORR_BASE + 5];
    o[0] = (lx + ly + lw + lh + loss_conf + loss_cls) / nGT_f;
  }
}

// ---------------- host entry -------------------------------------------------
extern "C" void kernel_launch(void* const* d_in, const int* in_sizes, int n_in,
                              void* d_out, int out_size, void* d_ws, size_t ws_size,
                              hipStream_t stream) {
  const float* out = (const float*)d_in[0];   // (16,125,128,128) fp32
  const float* tgt = (const float*)d_in[1];   // (16,250) fp32
  float* ws = (float*)d_ws;
  float* o  = (float*)d_out;

  int nblk = 512;                              // dense partials (<= PART_MAX)
  size_t need_bytes = (size_t)(CORR_BASE + 8) * sizeof(float);
  if (ws_size < need_bytes && ws_size >= (size_t)(PART_BASE + 72) * sizeof(float))
    nblk = 64;                                 // degrade gracefully if ws is tiny

  bt_kernel<<<1, 32, 0, stream>>>(tgt, ws);
  conf_dense_kernel<<<nblk, 256, 0, stream>>>(out, ws);
  corr_kernel<<<1, 256, 0, stream>>>(out, ws);
  finalize_kernel<<<1, 32, 0, stream>>>(ws, o, nblk);
}